// LVC_45792941310578
// MI455X (gfx1250) — compile-verified
//
#include <hip/hip_runtime.h>
#include <hip/hip_bf16.h>

typedef float v2f __attribute__((ext_vector_type(2)));
typedef float v8f __attribute__((ext_vector_type(8)));

#define DFEAT 128
#define LDSPAD 4
#define AROW (DFEAT + LDSPAD)   // 132 floats per staged row -> bank-conflict-free ds_load_b64

// ---------------- elementwise: out = (1 + eps[l]) * x ----------------
__global__ void lvc_scale_kernel(const float* __restrict__ x,
                                 const float* __restrict__ eps, int l,
                                 float* __restrict__ out, int n4) {
    int i = blockIdx.x * blockDim.x + threadIdx.x;
    if (i < n4) {
        float s = 1.0f + eps[l];
        float4 v = ((const float4*)x)[i];
        v.x *= s; v.y *= s; v.z *= s; v.w *= s;
        ((float4*)out)[i] = v;
    }
}

// ---------------- copy: h = x ----------------
__global__ void lvc_copy_kernel(const float* __restrict__ src,
                                float* __restrict__ dst, int n4) {
    int i = blockIdx.x * blockDim.x + threadIdx.x;
    if (i < n4) ((float4*)dst)[i] = ((const float4*)src)[i];
}

// ---------------- scatter: h[nidx[e]] += x[sidx[e]] ----------------
// one wave32 per edge; each lane covers 4 consecutive floats (32*4 = 128)
__global__ void lvc_scatter_kernel(const float* __restrict__ x,
                                   float* __restrict__ h,
                                   const int* __restrict__ sidx,
                                   const int* __restrict__ nidx, int E) {
    int lane = threadIdx.x & 31;
    int wave = (blockIdx.x * blockDim.x + threadIdx.x) >> 5;
    int nwaves = (gridDim.x * blockDim.x) >> 5;
    for (int e = wave; e < E; e += nwaves) {
        if (lane == 0 && (e + nwaves) < E) {
            __builtin_prefetch(&sidx[e + nwaves], 0, 0);   // global_prefetch_b8
            __builtin_prefetch(&nidx[e + nwaves], 0, 0);
        }
        int src = sidx[e];
        int dst = nidx[e];
        float4 v = ((const float4*)(x + (size_t)src * DFEAT))[lane];
        float* hd = h + (size_t)dst * DFEAT + lane * 4;
        __hip_atomic_fetch_add(hd + 0, v.x, __ATOMIC_RELAXED, __HIP_MEMORY_SCOPE_AGENT);
        __hip_atomic_fetch_add(hd + 1, v.y, __ATOMIC_RELAXED, __HIP_MEMORY_SCOPE_AGENT);
        __hip_atomic_fetch_add(hd + 2, v.z, __ATOMIC_RELAXED, __HIP_MEMORY_SCOPE_AGENT);
        __hip_atomic_fetch_add(hd + 3, v.w, __ATOMIC_RELAXED, __HIP_MEMORY_SCOPE_AGENT);
    }
}

// ---------------- GEMM accumulate: C[N,128] += A[N,128] @ B[128,128] ----------------
// blockDim = 256 (8 waves). Wave w owns column tile w (16 cols).
// B fragments for the whole K=128 kept in registers (32 x v2f = 64 VGPRs/lane).
// Per block iteration: 2 row tiles (32 rows) of A are staged once into LDS,
// then every wave reads fragments via ds_load_b64 (A fetched 1x from L2, not 8x).
// Two independent accumulator chains per wave -> 2x WMMA ILP.
__global__ __launch_bounds__(256) void lvc_gemm128_acc_kernel(
        const float* __restrict__ A, const float* __restrict__ B,
        float* __restrict__ C, int nRowTiles) {
    __shared__ __align__(16) float As[2 * 16 * AROW];

    int tid  = threadIdx.x;
    int lane = tid & 31;
    int wave = tid >> 5;                 // 0..7 -> column tile
    int colBase = wave * 16;
    int hi = lane >> 4;                  // 0: K pair {k,k+1}, 1: {k+2,k+3}
    int lo = lane & 15;                  // row (A) / col (B,C) within tile

    v2f bfrag[32];
#pragma unroll
    for (int s = 0; s < 32; ++s) {
        int k = 4 * s + 2 * hi;
        bfrag[s].x = B[(size_t)(k + 0) * DFEAT + colBase + lo];
        bfrag[s].y = B[(size_t)(k + 1) * DFEAT + colBase + lo];
    }

    for (int t0 = 2 * blockIdx.x; t0 < nRowTiles; t0 += 2 * gridDim.x) {
        bool two = (t0 + 1) < nRowTiles;          // block-uniform
        __syncthreads();                          // previous iteration done with LDS
        // stage 32 rows x 128 floats (1024 float4) with 256 threads -> 4 each
#pragma unroll
        for (int j = 0; j < 4; ++j) {
            int idx = tid + j * 256;              // float4 index 0..1023
            int row = idx >> 5;                   // 0..31 (16..31 = second tile)
            int c4  = idx & 31;
            float4 v = {0.0f, 0.0f, 0.0f, 0.0f};
            if (row < 16 || two) {
                size_t gr = (size_t)t0 * 16 + row;
                v = ((const float4*)(A + gr * DFEAT))[c4];
            }
            *(float4*)(&As[row * AROW + c4 * 4]) = v;   // ds_store_b128
        }
        __syncthreads();

        float* C0 = C + (size_t)t0 * 16 * DFEAT;
        float* C1 = C0 + 16 * DFEAT;
        v8f acc0, acc1 = {};
#pragma unroll
        for (int r = 0; r < 8; ++r)
            acc0[r] = C0[(size_t)(r + 8 * hi) * DFEAT + colBase + lo];
        if (two) {
#pragma unroll
            for (int r = 0; r < 8; ++r)
                acc1[r] = C1[(size_t)(r + 8 * hi) * DFEAT + colBase + lo];
        }
#pragma unroll
        for (int s = 0; s < 32; ++s) {
            int k = 4 * s + 2 * hi;
            v2f a0, a1;
            a0.x = As[lo * AROW + k];                     // ds_load_b64
            a0.y = As[lo * AROW + k + 1];
            a1.x = As[(lo + 16) * AROW + k];
            a1.y = As[(lo + 16) * AROW + k + 1];
            acc0 = __builtin_amdgcn_wmma_f32_16x16x4_f32(
                false, a0, false, bfrag[s], (short)0, acc0, false, false);
            acc1 = __builtin_amdgcn_wmma_f32_16x16x4_f32(
                false, a1, false, bfrag[s], (short)0, acc1, false, false);
        }
#pragma unroll
        for (int r = 0; r < 8; ++r)
            C0[(size_t)(r + 8 * hi) * DFEAT + colBase + lo] = acc0[r];
        if (two) {
#pragma unroll
            for (int r = 0; r < 8; ++r)
                C1[(size_t)(r + 8 * hi) * DFEAT + colBase + lo] = acc1[r];
        }
    }
}

// ---------------- head: Out[N,C] = A[N,128] @ Wh[128,C] + bh ----------------
// blockDim = 32 * nColTiles (C=40 -> 3 waves). Predication only on loads/stores;
// WMMA runs at full EXEC.
__global__ void lvc_head_kernel(const float* __restrict__ A,
                                const float* __restrict__ Wh,
                                const float* __restrict__ bh,
                                float* __restrict__ Out, int nRowTiles, int Cdim) {
    int lane = threadIdx.x & 31;
    int wave = threadIdx.x >> 5;        // column tile
    int colBase = wave * 16;
    int hi = lane >> 4;
    int lo = lane & 15;
    int col = colBase + lo;
    bool valid = col < Cdim;

    v2f bfrag[32];
#pragma unroll
    for (int s = 0; s < 32; ++s) {
        int k = 4 * s + 2 * hi;
        bfrag[s].x = valid ? Wh[(size_t)(k + 0) * Cdim + col] : 0.0f;
        bfrag[s].y = valid ? Wh[(size_t)(k + 1) * Cdim + col] : 0.0f;
    }
    float bias = valid ? bh[col] : 0.0f;

    for (int t = blockIdx.x; t < nRowTiles; t += gridDim.x) {
        const float* Arow = A + (size_t)t * 16 * DFEAT;
        v8f acc = {};
#pragma unroll
        for (int s = 0; s < 32; ++s) {
            int k = 4 * s + 2 * hi;
            v2f a;
            a.x = Arow[(size_t)lo * DFEAT + k + 0];
            a.y = Arow[(size_t)lo * DFEAT + k + 1];
            acc = __builtin_amdgcn_wmma_f32_16x16x4_f32(
                false, a, false, bfrag[s], (short)0, acc, false, false);
        }
        if (valid) {
#pragma unroll
            for (int r = 0; r < 8; ++r)
                Out[((size_t)t * 16 + r + 8 * hi) * Cdim + col] = acc[r] + bias;
        }
    }
}

extern "C" void kernel_launch(void* const* d_in, const int* in_sizes, int n_in,
                              void* d_out, int out_size, void* d_ws, size_t ws_size,
                              hipStream_t stream) {
    const float* x    = (const float*)d_in[0];  // [N,128]
    const float* Wmlp = (const float*)d_in[1];  // [L,HOPS,128,128]
    const float* eps  = (const float*)d_in[2];  // [L]
    const float* Wh   = (const float*)d_in[3];  // [128,C]
    const float* bh   = (const float*)d_in[4];  // [C]
    const int*   sidx = (const int*)d_in[5];    // [HOPS,E]
    const int*   nidx = (const int*)d_in[6];    // [HOPS,E]

    const int N    = in_sizes[0] / DFEAT;
    const int L    = in_sizes[2];
    const int Cdim = in_sizes[4];
    const int HOPS = in_sizes[1] / (L * DFEAT * DFEAT);
    const int E    = in_sizes[5] / HOPS;

    const int n4        = N * DFEAT / 4;
    const int nRowTiles = (N + 15) / 16;     // N=100000 -> 6250 exact

    float* buf0 = (float*)d_ws;
    float* buf1 = buf0 + (size_t)N * DFEAT;
    float* bufH = buf1 + (size_t)N * DFEAT;

    const int EW_BLK = 256;
    const int ewGrid = (n4 + EW_BLK - 1) / EW_BLK;
    const int scGrid = 4096;                 // grid-stride waves over edges
    const int gmGrid = 1024;                 // grid-stride over row-tile pairs
    const int nColTilesHead = (Cdim + 15) / 16;

    const float* xcur = x;
    float* outbufs[2] = { buf0, buf1 };

    for (int l = 0; l < L; ++l) {
        float* outl = outbufs[l & 1];
        lvc_scale_kernel<<<ewGrid, EW_BLK, 0, stream>>>(xcur, eps, l, outl, n4);
        for (int hop = 0; hop < HOPS; ++hop) {
            lvc_copy_kernel<<<ewGrid, EW_BLK, 0, stream>>>(xcur, bufH, n4);
            lvc_scatter_kernel<<<scGrid, EW_BLK, 0, stream>>>(
                xcur, bufH, sidx + (size_t)hop * E, nidx + (size_t)hop * E, E);
            lvc_gemm128_acc_kernel<<<gmGrid, 256, 0, stream>>>(
                bufH, Wmlp + ((size_t)l * HOPS + hop) * DFEAT * DFEAT, outl, nRowTiles);
        }
        xcur = outl;
    }
    lvc_head_kernel<<<gmGrid, 32 * nColTilesHead, 0, stream>>>(
        xcur, Wh, bh, (float*)d_out, nRowTiles, Cdim);
}